// LaBraMVectorQuantizer_28887950033222
// MI455X (gfx1250) — compile-verified
//
#include <hip/hip_runtime.h>

typedef float v2f __attribute__((ext_vector_type(2)));
typedef float v4f __attribute__((ext_vector_type(4)));
typedef float v8f __attribute__((ext_vector_type(8)));

#define N_TOK   16384   // B*S
#define D_MODEL 768
#define D_CODE  64
#define K_CODES 8192

__device__ __forceinline__ v8f wmma4(v2f a, v2f b, v8f c) {
  // D = A(16x4,f32) * B(4x16,f32) + C(16x16,f32)
  return __builtin_amdgcn_wmma_f32_16x16x4_f32(
      /*neg_a=*/false, a, /*neg_b=*/false, b,
      /*c_mod=*/(short)0, c, /*reuse_a=*/false, /*reuse_b=*/false);
}

// -------- Kernel 1: H = tanh(Z @ W1 + b1), (16384x768)x(768x768) --------
// Register-blocked 4 row-tiles x 2 col-tiles per wave: 8 VMEM per 8 WMMA.
__global__ __launch_bounds__(128) void k_gemm1_tanh(
    const float* __restrict__ Z, const float* __restrict__ W1,
    const float* __restrict__ b1, float* __restrict__ H) {
  const int wave = blockIdx.x * 4 + (threadIdx.x >> 5);
  const int lane = threadIdx.x & 31;
  const int hf = lane >> 4, m = lane & 15;
  const int NCG = D_MODEL / 32;          // 24 col groups (2 tiles of 16)
  const int cg = wave % NCG;
  const int rg = wave / NCG;             // 256 row groups (64 rows)
  const int col0 = cg * 32 + m;

  v8f acc[4][2];
#pragma unroll
  for (int t = 0; t < 4; ++t) { acc[t][0] = (v8f){}; acc[t][1] = (v8f){}; }

  const float* a0 = Z  + (size_t)(rg * 64 + m) * D_MODEL + 2 * hf;
  const float* b0 = W1 + (size_t)(2 * hf) * D_MODEL + col0;
  for (int k = 0; k < D_MODEL; k += 4) {
    const float* bk = b0 + (size_t)k * D_MODEL;
    v2f bA, bB;
    bA.x = bk[0];  bA.y = bk[D_MODEL];
    bB.x = bk[16]; bB.y = bk[D_MODEL + 16];
#pragma unroll
    for (int t = 0; t < 4; ++t) {
      v2f a = *(const v2f*)(a0 + (size_t)(t * 16) * D_MODEL + k);
      acc[t][0] = wmma4(a, bA, acc[t][0]);
      acc[t][1] = wmma4(a, bB, acc[t][1]);
    }
  }
  const float bias0 = b1[col0], bias1 = b1[col0 + 16];
#pragma unroll
  for (int t = 0; t < 4; ++t) {
    float* orow = H + (size_t)(rg * 64 + t * 16 + 8 * hf) * D_MODEL + col0;
#pragma unroll
    for (int r = 0; r < 8; ++r) {
      orow[(size_t)r * D_MODEL]      = tanhf(acc[t][0][r] + bias0);
      orow[(size_t)r * D_MODEL + 16] = tanhf(acc[t][1][r] + bias1);
    }
  }
}

// -------- Kernel 2: Ze = l2norm(H @ W2 + b2), rows of 64 --------
__global__ __launch_bounds__(128) void k_gemm2_norm(
    const float* __restrict__ H, const float* __restrict__ W2,
    const float* __restrict__ b2, float* __restrict__ Ze) {
  __shared__ float ssq[16];
  const int wv = threadIdx.x >> 5;        // 4 waves -> 4 col tiles of 16
  const int lane = threadIdx.x & 31;
  const int hf = lane >> 4, m = lane & 15;
  const int row0 = blockIdx.x * 16;
  const int col = wv * 16 + m;

  if (threadIdx.x < 16) ssq[threadIdx.x] = 0.f;
  __syncthreads();

  v8f acc = {};
  const float* arow = H  + (size_t)(row0 + m) * D_MODEL + 2 * hf;
  const float* bcol = W2 + (size_t)(2 * hf) * D_CODE + col;
  for (int k = 0; k < D_MODEL; k += 4) {
    v2f a = *(const v2f*)(arow + k);
    v2f b;
    b.x = bcol[(size_t)k * D_CODE];
    b.y = bcol[(size_t)k * D_CODE + D_CODE];
    acc = wmma4(a, b, acc);
  }
  const float bias = b2[col];
#pragma unroll
  for (int r = 0; r < 8; ++r) acc[r] += bias;

  // per-row sum of squares: reduce 16 lanes of each half, accumulate in LDS
#pragma unroll
  for (int r = 0; r < 8; ++r) {
    float v = acc[r] * acc[r];
    for (int msk = 8; msk >= 1; msk >>= 1) v += __shfl_xor(v, msk, 32);
    if (m == 0) atomicAdd(&ssq[r + 8 * hf], v);
  }
  __syncthreads();
#pragma unroll
  for (int r = 0; r < 8; ++r) {
    const int row = r + 8 * hf;
    const float s = 1.f / fmaxf(sqrtf(ssq[row]), 1e-12f);
    Ze[(size_t)(row0 + row) * D_CODE + col] = acc[r] * s;
  }
}

// -------- Kernel 3: sim = Ze @ emb^T, argmax_k, one-hot, commit loss --------
// emb chunks are streamed L2 -> LDS with async (ASYNCcnt) loads, double-buffered.
__global__ __launch_bounds__(256) void k_sim_argmax(
    const float* __restrict__ Ze, const float* __restrict__ emb,
    float* __restrict__ probs, int* __restrict__ idxbuf, float* __restrict__ loss) {
  __shared__ float embS[2][16 * 64];     // 2 x 4KB chunk buffers
  const int wv = threadIdx.x >> 5;       // 0..7, each wave owns 16 token rows
  const int lane = threadIdx.x & 31;
  const int hf = lane >> 4, m = lane & 15;
  const int row0 = (blockIdx.x * 8 + wv) * 16;

  // preload A fragments: Ze rows row0..row0+15, K=64 (16 k-steps)
  v2f afr[16];
#pragma unroll
  for (int s = 0; s < 16; ++s)
    afr[s] = *(const v2f*)(Ze + (size_t)(row0 + m) * D_CODE + 4 * s + 2 * hf);

  float bestv[8]; int besti[8];
#pragma unroll
  for (int r = 0; r < 8; ++r) { bestv[r] = -3.4e38f; besti[r] = 0; }

  // LDS byte offset of the staging buffers (generic shared ptr low 32 bits)
  const unsigned ldsbase = (unsigned)(uintptr_t)(&embS[0][0]);
  const unsigned tb = threadIdx.x * 16u;  // 256 lanes x b128 = 4KB chunk

  // prologue: async-load chunk 0 into buffer 0
  {
    unsigned dst = ldsbase + tb;
    unsigned src = tb;
    asm volatile("global_load_async_to_lds_b128 %0, %1, %2"
                 :: "v"(dst), "v"(src), "s"(emb) : "memory");
  }

  const int NCHUNK = K_CODES / 16;       // 512
  for (int t = 0; t < NCHUNK; ++t) {
    asm volatile("s_wait_asynccnt 0x0" ::: "memory");  // my chunk-t load done
    __syncthreads();                                   // everyone's done; alt buf free
    if (t + 1 < NCHUNK) {                              // prefetch chunk t+1
      unsigned dst = ldsbase + ((unsigned)(t + 1) & 1u) * 4096u + tb;
      unsigned src = (unsigned)(t + 1) * 4096u + tb;
      asm volatile("global_load_async_to_lds_b128 %0, %1, %2"
                   :: "v"(dst), "v"(src), "s"(emb) : "memory");
    }
    const float* bb = &embS[t & 1][0];
    v8f acc = {};
#pragma unroll
    for (int s = 0; s < 16; ++s) {
      v2f b = *(const v2f*)(bb + m * 64 + 4 * s + 2 * hf); // emb[c0+m][k] = B[k][m]
      acc = wmma4(afr[s], b, acc);
    }
    const int c0 = t * 16;
#pragma unroll
    for (int r = 0; r < 8; ++r)
      if (acc[r] > bestv[r]) { bestv[r] = acc[r]; besti[r] = c0 + m; }
  }

  // reduce (val,idx) across the 16 lanes of each half; ties -> lowest index
#pragma unroll
  for (int r = 0; r < 8; ++r) {
    float v = bestv[r]; int i = besti[r];
    for (int msk = 8; msk >= 1; msk >>= 1) {
      float ov = __shfl_xor(v, msk, 32);
      int   oi = __shfl_xor(i, msk, 32);
      if (ov > v || (ov == v && oi < i)) { v = ov; i = oi; }
    }
    bestv[r] = v; besti[r] = i;
  }

  if (m == 0) {
#pragma unroll
    for (int r = 0; r < 8; ++r) {
      const int row = row0 + r + 8 * hf;
      probs[(size_t)row * K_CODES + besti[r]] = 1.0f;
      idxbuf[row] = besti[r];
    }
  }

  // commitment loss: each lane owns row m's K-slice; fetch that row's index
  const int rsel = m & 7;
  const int src = (m < 8) ? m : (m + 8);  // lane whose besti[rsel] is row m
  int myidx = besti[0];
#pragma unroll
  for (int r = 1; r < 8; ++r) if (rsel == r) myidx = besti[r];
  const int idx_m = __shfl(myidx, src, 32);

  const float* qrow = emb + (size_t)idx_m * D_CODE + 2 * hf;
  float lsum = 0.f;
#pragma unroll
  for (int s = 0; s < 16; ++s) {
    v2f q = *(const v2f*)(qrow + 4 * s);
    float d0 = q.x - afr[s].x, d1 = q.y - afr[s].y;
    lsum += d0 * d0 + d1 * d1;
  }
  for (int msk = 16; msk >= 1; msk >>= 1) lsum += __shfl_xor(lsum, msk, 32);
  if (lane == 0) atomicAdd(loss, lsum * (1.0f / (float)(N_TOK * D_CODE)));
}

// -------- Kernel 4: out = LayerNorm(emb[idx]) @ Wp + bp --------
__global__ __launch_bounds__(128) void k_ln_gemm3(
    const int* __restrict__ idxbuf, const float* __restrict__ emb,
    const float* __restrict__ gamma, const float* __restrict__ beta,
    const float* __restrict__ Wp, const float* __restrict__ bp,
    float* __restrict__ out) {
  __shared__ float Aln[16 * 64];
  const int row0 = blockIdx.x * 16;

  if (threadIdx.x < 16) {
    const int row = threadIdx.x;
    const float* q = emb + (size_t)idxbuf[row0 + row] * D_CODE;
    float s = 0.f, s2 = 0.f;
#pragma unroll
    for (int c = 0; c < 64; ++c) { float x = q[c]; s += x; s2 += x * x; }
    const float mu = s * (1.f / 64.f);
    const float var = s2 * (1.f / 64.f) - mu * mu;
    const float rs = rsqrtf(var + 1e-5f);
#pragma unroll
    for (int c = 0; c < 64; ++c)
      Aln[row * 64 + c] = (q[c] - mu) * rs * gamma[c] + beta[c];
  }
  __syncthreads();

  const int wv = threadIdx.x >> 5, lane = threadIdx.x & 31;
  const int hf = lane >> 4, m = lane & 15;
  v2f afr[16];
#pragma unroll
  for (int s = 0; s < 16; ++s)
    afr[s] = *(const v2f*)(Aln + m * 64 + 4 * s + 2 * hf);

  for (int ct = wv; ct < D_MODEL / 16; ct += 4) {
    const int col = ct * 16 + m;
    v8f acc = {};
    const float* bcol = Wp + (size_t)(2 * hf) * D_MODEL + col;
#pragma unroll
    for (int s = 0; s < 16; ++s) {
      v2f b;
      b.x = bcol[(size_t)(4 * s) * D_MODEL];
      b.y = bcol[(size_t)(4 * s + 1) * D_MODEL];
      acc = wmma4(afr[s], b, acc);
    }
    const float bias = bp[col];
#pragma unroll
    for (int r = 0; r < 8; ++r)
      out[(size_t)(row0 + r + 8 * hf) * D_MODEL + col] = acc[r] + bias;
  }
}

extern "C" void kernel_launch(void* const* d_in, const int* in_sizes, int n_in,
                              void* d_out, int out_size, void* d_ws, size_t ws_size,
                              hipStream_t stream) {
  const float* Z   = (const float*)d_in[0];
  const float* W1  = (const float*)d_in[1];
  const float* b1  = (const float*)d_in[2];
  const float* W2  = (const float*)d_in[3];
  const float* b2  = (const float*)d_in[4];
  const float* emb = (const float*)d_in[5];
  const float* g   = (const float*)d_in[6];
  const float* be  = (const float*)d_in[7];
  const float* Wp  = (const float*)d_in[8];
  const float* bp  = (const float*)d_in[9];

  float* out   = (float*)d_out;                 // (N,768)
  float* loss  = out + (size_t)N_TOK * D_MODEL; // scalar
  float* probs = loss + 1;                      // (N,8192)

  float* H   = (float*)d_ws;                          // 48 MB
  float* Ze  = H + (size_t)N_TOK * D_MODEL;           // 4 MB
  int*   idx = (int*)(Ze + (size_t)N_TOK * D_CODE);   // 64 KB

  // one-hot is ~all of the memory traffic: do it as one wide memset (+ loss=0)
  hipMemsetAsync(loss, 0, (1 + (size_t)N_TOK * K_CODES) * sizeof(float), stream);

  k_gemm1_tanh<<<(N_TOK / 64) * (D_MODEL / 32) / 4, 128, 0, stream>>>(Z, W1, b1, H);
  k_gemm2_norm<<<N_TOK / 16, 128, 0, stream>>>(H, W2, b2, Ze);
  k_sim_argmax<<<N_TOK / 128, 256, 0, stream>>>(Ze, emb, probs, idx, loss);
  k_ln_gemm3<<<N_TOK / 16, 128, 0, stream>>>(idx, emb, g, be, Wp, bp, out);
}